// CategoricalProjection_31877247271153
// MI455X (gfx1250) — compile-verified
//
#include <hip/hip_runtime.h>
#include <hip/hip_bf16.h>
#include <stdint.h>

// C51 categorical projection for MI455X (gfx1250).
// Memory-bound: ~218 MB of traffic -> ~9.4 us at 23.3 TB/s HBM.
// Strategy:
//   - TDM tensor_load_to_lds DMAs a [128 x 51] f32 probs tile into LDS.
//   - Each thread owns one row; scatters into a private 51-float LDS
//     accumulator row via single-instruction ds_add_f32 atomics.
//   - TDM tensor_store_from_lds DMAs the finished accumulator tile back to
//     global memory (S_ENDPGM's implicit wait-idle drains TENSORcnt, keeping
//     the workgroup's LDS alive until the DMA completes).

#define NUM_ATOMS   51
#define ROWS        128            // rows (== threads) per block, 4 waves
#define V_MIN_F     (-10.0f)
#define V_MAX_F     (10.0f)
#define ATOM_DELTA  0.4f
#define DISCOUNT_F  0.99f

typedef unsigned int u32x4 __attribute__((ext_vector_type(4)));
typedef int          i32x4 __attribute__((ext_vector_type(4)));
typedef int          i32x8 __attribute__((ext_vector_type(8)));

// Build D# groups for a 2-D [ROWS x NUM_ATOMS] f32 tile at `gaddr` <-> `lds_off`.
__device__ __forceinline__ void make_tile_desc(unsigned long long gaddr,
                                               unsigned lds_off,
                                               u32x4& g0, i32x8& g1)
{
    g0[0] = 1u;                                      // count=1 (valid), no gather
    g0[1] = lds_off;                                 // lds_addr (bytes)
    g0[2] = (unsigned)(gaddr & 0xFFFFFFFFu);         // global_addr[31:0]
    g0[3] = (unsigned)((gaddr >> 32) & 0x1FFFFFFu)   // global_addr[56:32]
          | (2u << 30);                              // type = 2 ("image")

    g1[0] = (int)(2u << 16);                         // wg_mask=0, data_size=2 (4 B)
    g1[1] = (int)((unsigned)NUM_ATOMS << 16);        // tensor_dim0[15:0] @ bits 63:48
    g1[2] = (int)((unsigned)ROWS << 16);             // tensor_dim1[15:0] @ bits 95:80
    g1[3] = (int)((unsigned)NUM_ATOMS << 16);        // tile_dim0 @ bits 127:112
    g1[4] = (int)ROWS;                               // tile_dim1 @ bits 143:128
    g1[5] = (int)NUM_ATOMS;                          // tensor_dim0_stride[31:0]
    g1[6] = 0;
    g1[7] = 0;
}

__global__ __launch_bounds__(ROWS) void c51_project_kernel(
    const float* __restrict__ reward,
    const float* __restrict__ probs,
    const float* __restrict__ not_done,
    float* __restrict__ out)
{
    __shared__ __align__(16) float stage[ROWS * NUM_ATOMS];  // 26,112 B
    __shared__ __align__(16) float acc[ROWS * NUM_ATOMS];    // 26,112 B

    const int tid = threadIdx.x;
    const long long rowBase = (long long)blockIdx.x * ROWS;

    // ---- 1. Wave 0 issues the TDM DMA: probs tile -> LDS stage.
    //         (TDM ignores EXEC; wave-uniform branch keeps other waves out.)
    if (tid < 32) {
        u32x4 g0; i32x8 g1;
        make_tile_desc((unsigned long long)(const void*)(probs + rowBase * NUM_ATOMS),
                       (unsigned)(unsigned long long)(const void*)&stage[0], g0, g1);
        i32x4 g2 = {0, 0, 0, 0};                     // 2-D tensor: groups 2/3 unused
        i32x4 g3 = {0, 0, 0, 0};
        i32x8 g4 = {0, 0, 0, 0, 0, 0, 0, 0};
        __builtin_amdgcn_tensor_load_to_lds(g0, g1, g2, g3, g4, 0);
    }

    // ---- 2. Zero the accumulator while the DMA is in flight (b128 stores).
    {
        float4* accv = (float4*)acc;
        const float4 z = make_float4(0.f, 0.f, 0.f, 0.f);
        for (int i = tid; i < (ROWS * NUM_ATOMS) / 4; i += ROWS) accv[i] = z;
    }

    // ---- 3. Coalesced scalar-per-row loads.
    const float r  = reward[rowBase + tid];
    const float nd = not_done[rowBase + tid];

    __builtin_amdgcn_s_wait_tensorcnt(0);   // no-op for waves with TENSORcnt==0
    __syncthreads();

    // ---- 4. Per-row projection; scatter via single-instruction ds_add_f32.
    //         Lane L's row base word is L*51; gcd(51,64)==1 so the 32 lane
    //         bases land in 32 distinct banks.
    const float* prow = &stage[tid * NUM_ATOMS];
    float*       arow = &acc[tid * NUM_ATOMS];
    const float  dn   = DISCOUNT_F * nd;

#pragma unroll
    for (int j = 0; j < NUM_ATOMS; ++j) {
        float av = V_MIN_F + ATOM_DELTA * (float)j;
        float v  = fmaf(dn, av, r);
        v        = fminf(fmaxf(v, V_MIN_F), V_MAX_F);
        float b  = (v - V_MIN_F) / ATOM_DELTA;           // in [0, 50]
        float l  = floorf(b);
        float u  = ceilf(b);
        float pj = prow[j];
        float eq = (l == u) ? 1.0f : 0.0f;
        float dml = (u + eq - b) * pj;
        float dmu = (b - l) * pj;
        atomicAdd(&arow[(int)l], dml);                   // ds_add_f32 (no rtn)
        atomicAdd(&arow[(int)u], dmu);                   // ds_add_f32 (no rtn)
    }

    __syncthreads();   // all ds_add_f32 complete & visible across the workgroup

    // ---- 5. Wave 0 issues the TDM DMA: LDS acc tile -> out.
    //         S_ENDPGM's implicit wait-idle holds wave 0 (and thus the
    //         workgroup's LDS allocation) until TENSORcnt reaches zero.
    if (tid < 32) {
        u32x4 g0; i32x8 g1;
        make_tile_desc((unsigned long long)(const void*)(out + rowBase * NUM_ATOMS),
                       (unsigned)(unsigned long long)(const void*)&acc[0], g0, g1);
        i32x4 g2 = {0, 0, 0, 0};
        i32x4 g3 = {0, 0, 0, 0};
        i32x8 g4 = {0, 0, 0, 0, 0, 0, 0, 0};
        __builtin_amdgcn_tensor_store_from_lds(g0, g1, g2, g3, g4, 0);
    }
}

// Defensive tail path (BS=524288 is divisible by 128, so normally unused).
__global__ __launch_bounds__(ROWS) void c51_project_tail_kernel(
    const float* __restrict__ reward,
    const float* __restrict__ probs,
    const float* __restrict__ not_done,
    float* __restrict__ out,
    long long row0, int nrows)
{
    __shared__ float acc[ROWS * NUM_ATOMS];
    const int tid = threadIdx.x;
    for (int i = tid; i < ROWS * NUM_ATOMS; i += ROWS) acc[i] = 0.f;
    __syncthreads();
    if (tid < nrows) {
        const long long row = row0 + tid;
        const float r  = reward[row];
        const float dn = DISCOUNT_F * not_done[row];
        const float* prow = probs + row * NUM_ATOMS;
        float* arow = &acc[tid * NUM_ATOMS];
        for (int j = 0; j < NUM_ATOMS; ++j) {
            float av = V_MIN_F + ATOM_DELTA * (float)j;
            float v  = fmaf(dn, av, r);
            v        = fminf(fmaxf(v, V_MIN_F), V_MAX_F);
            float b  = (v - V_MIN_F) / ATOM_DELTA;
            float l  = floorf(b);
            float u  = ceilf(b);
            float pj = prow[j];
            float eq = (l == u) ? 1.0f : 0.0f;
            atomicAdd(&arow[(int)l], (u + eq - b) * pj);
            atomicAdd(&arow[(int)u], (b - l) * pj);
        }
    }
    __syncthreads();
    for (int i = tid; i < nrows * NUM_ATOMS; i += ROWS)
        out[row0 * NUM_ATOMS + i] = acc[i];
}

extern "C" void kernel_launch(void* const* d_in, const int* in_sizes, int n_in,
                              void* d_out, int out_size, void* d_ws, size_t ws_size,
                              hipStream_t stream) {
    (void)n_in; (void)out_size; (void)d_ws; (void)ws_size;
    const float* reward   = (const float*)d_in[0];
    const float* probs    = (const float*)d_in[1];
    const float* not_done = (const float*)d_in[2];
    float* out = (float*)d_out;

    const int bs      = in_sizes[0];       // reward is [BS, 1]
    const int nblocks = bs / ROWS;
    const int rem     = bs - nblocks * ROWS;

    if (nblocks > 0) {
        c51_project_kernel<<<nblocks, ROWS, 0, stream>>>(reward, probs, not_done, out);
    }
    if (rem > 0) {
        c51_project_tail_kernel<<<1, ROWS, 0, stream>>>(
            reward, probs, not_done, out, (long long)nblocks * ROWS, rem);
    }
}